// ScalarDotProductCriticNetworkV5_78993038508710
// MI455X (gfx1250) — compile-verified
//
#include <hip/hip_runtime.h>
#include <hip/hip_bf16.h>
#include <math.h>

typedef __attribute__((ext_vector_type(2))) float v2f;
typedef __attribute__((ext_vector_type(8))) float v8f;

namespace {

constexpr int kN    = 32;   // agents
constexpr int kDobs = 120;
constexpr int kAct  = 8;
constexpr int kDin  = 128;

// ---- LDS layout (float offsets), padded strides for bank-conflict-free rows ----
constexpr int LDSTATE = 124;  // 120 + 4
constexpr int LDSM    = 12;   // 8 + 4
constexpr int LD64    = 68;   // 64 + 4
constexpr int LD32    = 36;   // 32 + 4

constexpr int OFF_STATE = 0;      // 32*124 = 3968   (later reused: S at [0,2176), SF at [2176,4352))
constexpr int OFF_ACT   = 3968;   // 32*12  = 384
constexpr int OFF_POL   = 4352;   // 384
constexpr int OFF_SEMB  = 4736;   // 32*68  = 2176   (reused: emb_act)
constexpr int OFF_Q     = 6912;   // 2176           (reused: emb_pol)
constexpr int OFF_K     = 9088;   // 2176           (reused: av_act, then DF)
constexpr int OFF_SC    = 11264;  // 32*36 = 1152
constexpr int OFF_W     = 12416;  // 1152           (weight, live to end)
constexpr int OFF_DELTA = 13568;  // 2176
constexpr int OFF_S     = 0;
constexpr int OFF_SF    = 2176;
constexpr int OFF_DF    = 9088;
constexpr int OFF_WF2   = 15744;  // 64
constexpr int SMEM_FLOATS = 15808; // 63232 bytes < 64KB

__device__ __forceinline__ v8f wmma_step(v8f c, v2f a, v2f b) {
  // V_WMMA_F32_16X16X4_F32 : D = A(16x4,f32) * B(4x16,f32) + C(16x16,f32)
  return __builtin_amdgcn_wmma_f32_16x16x4_f32(false, a, false, b, (short)0, c,
                                               false, false);
}

// A tile base: row-major [16][K] with stride lda (LDS).
// Bt tile base: row-major [16][K] = transposed B, stride ldb (global weight [n][k] or LDS).
// Fragment layout (ISA 7.12.2, 32-bit): lane<16 -> K={kk,kk+1}; lane>=16 -> K={kk+2,kk+3}.
__device__ __forceinline__ v8f acc_BT(v8f c, const float* __restrict__ A, int lda,
                                      const float* __restrict__ Bt, int ldb,
                                      int K, int lm, int half) {
#pragma unroll
  for (int kk = 0; kk < K; kk += 4) {
    const int k0 = kk + 2 * half;
    v2f a, bf;
    a.x  = A[lm * lda + k0];
    a.y  = A[lm * lda + k0 + 1];
    bf.x = Bt[lm * ldb + k0];
    bf.y = Bt[lm * ldb + k0 + 1];
    c = wmma_step(c, a, bf);
  }
  return c;
}

// B in natural row-major [K][N] form (e.g. an LDS activation matrix), tile base offset to col.
__device__ __forceinline__ v8f acc_BN(v8f c, const float* __restrict__ A, int lda,
                                      const float* __restrict__ B, int ldb,
                                      int K, int lm, int half) {
#pragma unroll
  for (int kk = 0; kk < K; kk += 4) {
    const int k0 = kk + 2 * half;
    v2f a, bf;
    a.x  = A[lm * lda + k0];
    a.y  = A[lm * lda + k0 + 1];
    bf.x = B[k0 * ldb + lm];
    bf.y = B[(k0 + 1) * ldb + lm];
    c = wmma_step(c, a, bf);
  }
  return c;
}

// C/D layout: VGPR r -> row r (lanes 0-15) / row r+8 (lanes 16-31), col = lane%16.
__device__ __forceinline__ void store_tile(float* D, int ldd, v8f c, int lm, int half) {
#pragma unroll
  for (int r = 0; r < 8; ++r) D[(r + 8 * half) * ldd + lm] = c[r];
}

__global__ __launch_bounds__(256) void critic_fused(
    const float* __restrict__ states, const float* __restrict__ policies,
    const float* __restrict__ actions,
    const float* __restrict__ W_obs, const float* __restrict__ b_obs,
    const float* __restrict__ W_oa, const float* __restrict__ b_oa,
    const float* __restrict__ W_k, const float* __restrict__ b_k,
    const float* __restrict__ W_q, const float* __restrict__ b_q,
    const float* __restrict__ W_av, const float* __restrict__ W_f1,
    const float* __restrict__ W_f2,
    float* __restrict__ out_value, float* __restrict__ out_weight) {
  __shared__ float sm[SMEM_FLOATS];
  const int b    = blockIdx.x;
  const int tid  = threadIdx.x;
  const int wv   = tid >> 5;       // wave id (wave32), 8 waves
  const int lane = tid & 31;
  const int lm   = lane & 15;
  const int half = lane >> 4;
  const int mi   = wv >> 2;        // 2x4 tile grid over 32x64 outputs
  const int ni   = wv & 3;
  const int col  = ni * 16 + lm;

  // ---------- phase 0: stage per-batch inputs ----------
  for (int idx = tid; idx < kN * kDobs; idx += 256) {
    const int r = idx / kDobs, c = idx % kDobs;
    sm[OFF_STATE + r * LDSTATE + c] = states[(size_t)b * kN * kDobs + idx];
  }
  for (int idx = tid; idx < kN * kAct; idx += 256) {
    const int r = idx / kAct, c = idx % kAct;
    sm[OFF_ACT + r * LDSM + c] = actions[(size_t)b * kN * kAct + idx];
    sm[OFF_POL + r * LDSM + c] = policies[(size_t)b * kN * kAct + idx];
  }
  if (tid < 64) sm[OFF_WF2 + tid] = W_f2[tid];
  __syncthreads();

  // ---------- phase 1: states_emb = relu(states @ W_obs^T + b_obs)  [32x64] ----------
  {
    v8f c = {};
    c = acc_BT(c, &sm[OFF_STATE + mi * 16 * LDSTATE], LDSTATE,
               W_obs + ni * 16 * kDobs, kDobs, kDobs, lm, half);
    const float bias = b_obs[col];
#pragma unroll
    for (int r = 0; r < 8; ++r) { float x = c[r] + bias; c[r] = x > 0.f ? x : 0.f; }
    store_tile(&sm[OFF_SEMB + mi * 16 * LD64 + ni * 16], LD64, c, lm, half);
  }
  __syncthreads();

  // ---------- phase 2: q = semb@W_q^T + b_q ; k = semb@W_k^T + b_k ----------
  {
    v8f cq = {}, ck = {};
    cq = acc_BT(cq, &sm[OFF_SEMB + mi * 16 * LD64], LD64, W_q + ni * 16 * 64, 64, 64, lm, half);
    ck = acc_BT(ck, &sm[OFF_SEMB + mi * 16 * LD64], LD64, W_k + ni * 16 * 64, 64, 64, lm, half);
    const float bq = b_q[col], bk = b_k[col];
#pragma unroll
    for (int r = 0; r < 8; ++r) { cq[r] += bq; ck[r] += bk; }
    store_tile(&sm[OFF_Q + mi * 16 * LD64 + ni * 16], LD64, cq, lm, half);
    store_tile(&sm[OFF_K + mi * 16 * LD64 + ni * 16], LD64, ck, lm, half);
  }
  __syncthreads();

  // ---------- phase 3: scores[i][j] = q_i . k_j  [32x32], waves 0-3 ----------
  if (wv < 4) {
    const int smi = wv >> 1, sni = wv & 1;
    v8f c = {};
    c = acc_BT(c, &sm[OFF_Q + smi * 16 * LD64], LD64,
               &sm[OFF_K + sni * 16 * LD64], LD64, 64, lm, half);
    store_tile(&sm[OFF_SC + smi * 16 * LD32 + sni * 16], LD32, c, lm, half);
  }
  __syncthreads();

  // ---------- phase 4: weight[j][i] = softmax_i(scores[i][j]/8); write output ----------
  if (tid < 32) {
    const int j = tid;
    float mx = -INFINITY;
    for (int i = 0; i < 32; ++i) mx = fmaxf(mx, sm[OFF_SC + i * LD32 + j]);
    float sum = 0.f;
    for (int i = 0; i < 32; ++i) {
      const float e = expf((sm[OFF_SC + i * LD32 + j] - mx) * 0.125f);
      sm[OFF_W + j * LD32 + i] = e;
      sum += e;
    }
    const float inv = 1.f / sum;
    for (int i = 0; i < 32; ++i) {
      const float v = sm[OFF_W + j * LD32 + i] * inv;
      sm[OFF_W + j * LD32 + i] = v;
      out_weight[(size_t)b * 1024 + j * 32 + i] = v;
    }
  }
  __syncthreads();

  // ---------- phase 5: emb_act/emb_pol = relu([states,act|pol]@W_oa^T + b_oa) ----------
  // obs part (K=120) shared between both; then K=8 tail with actions vs policies.
  {
    v8f cc = {};
    cc = acc_BT(cc, &sm[OFF_STATE + mi * 16 * LDSTATE], LDSTATE,
                W_oa + ni * 16 * kDin, kDin, kDobs, lm, half);
    v8f ca = cc, cp = cc;
    ca = acc_BT(ca, &sm[OFF_ACT + mi * 16 * LDSM], LDSM,
                W_oa + ni * 16 * kDin + kDobs, kDin, kAct, lm, half);
    cp = acc_BT(cp, &sm[OFF_POL + mi * 16 * LDSM], LDSM,
                W_oa + ni * 16 * kDin + kDobs, kDin, kAct, lm, half);
    const float bias = b_oa[col];
#pragma unroll
    for (int r = 0; r < 8; ++r) {
      float xa = ca[r] + bias; ca[r] = xa > 0.f ? xa : 0.f;
      float xp = cp[r] + bias; cp[r] = xp > 0.f ? xp : 0.f;
    }
    store_tile(&sm[OFF_SEMB + mi * 16 * LD64 + ni * 16], LD64, ca, lm, half); // emb_act
    store_tile(&sm[OFF_Q    + mi * 16 * LD64 + ni * 16], LD64, cp, lm, half); // emb_pol
  }
  __syncthreads();

  // ---------- phase 6: av_act = tanh(emb_act@W_av^T); delta = tanh(emb_pol@W_av^T) - av_act ----------
  {
    v8f c1 = {}, c2 = {};
    c1 = acc_BT(c1, &sm[OFF_SEMB + mi * 16 * LD64], LD64, W_av + ni * 16 * 64, 64, 64, lm, half);
    c2 = acc_BT(c2, &sm[OFF_Q    + mi * 16 * LD64], LD64, W_av + ni * 16 * 64, 64, 64, lm, half);
#pragma unroll
    for (int r = 0; r < 8; ++r) {
      const float t1 = tanhf(c1[r]);
      const float t2 = tanhf(c2[r]);
      c1[r] = t1;
      c2[r] = t2 - t1;
    }
    store_tile(&sm[OFF_K     + mi * 16 * LD64 + ni * 16], LD64, c1, lm, half); // av_act
    store_tile(&sm[OFF_DELTA + mi * 16 * LD64 + ni * 16], LD64, c2, lm, half); // delta
  }
  __syncthreads();

  // ---------- phase 7: S[a][d] = sum_i weight[a][i] * av_act[i][d]  [32x64] ----------
  {
    v8f c = {};
    c = acc_BN(c, &sm[OFF_W + mi * 16 * LD32], LD32, &sm[OFF_K + ni * 16], LD64,
               32, lm, half);
    store_tile(&sm[OFF_S + mi * 16 * LD64 + ni * 16], LD64, c, lm, half);
  }
  __syncthreads();

  // ---------- phase 8: SF = (S@W_f1^T)/N ; DF = (delta@W_f1^T)/N ----------
  {
    v8f c1 = {}, c2 = {};
    c1 = acc_BT(c1, &sm[OFF_S     + mi * 16 * LD64], LD64, W_f1 + ni * 16 * 64, 64, 64, lm, half);
    c2 = acc_BT(c2, &sm[OFF_DELTA + mi * 16 * LD64], LD64, W_f1 + ni * 16 * 64, 64, 64, lm, half);
#pragma unroll
    for (int r = 0; r < 8; ++r) { c1[r] *= 0.03125f; c2[r] *= 0.03125f; }
    store_tile(&sm[OFF_SF + mi * 16 * LD64 + ni * 16], LD64, c1, lm, half);
    store_tile(&sm[OFF_DF + mi * 16 * LD64 + ni * 16], LD64, c2, lm, half);
  }
  __syncthreads();

  // ---------- phase 9: value[a][c] = sum_h leaky(SF[a,h] + w[a,c]*DF[c,h]) * W_f2[h] ----------
  {
    const int a  = tid >> 3;
    const int cb = (tid & 7) * 4;
    const float w0 = sm[OFF_W + a * LD32 + cb + 0];
    const float w1 = sm[OFF_W + a * LD32 + cb + 1];
    const float w2 = sm[OFF_W + a * LD32 + cb + 2];
    const float w3 = sm[OFF_W + a * LD32 + cb + 3];
    float a0 = 0.f, a1 = 0.f, a2 = 0.f, a3 = 0.f;
#pragma unroll 4
    for (int h = 0; h < 64; ++h) {
      const float sf = sm[OFF_SF + a * LD64 + h];
      const float wf = sm[OFF_WF2 + h];
      float t0 = sf + w0 * sm[OFF_DF + (cb + 0) * LD64 + h];
      float t1 = sf + w1 * sm[OFF_DF + (cb + 1) * LD64 + h];
      float t2 = sf + w2 * sm[OFF_DF + (cb + 2) * LD64 + h];
      float t3 = sf + w3 * sm[OFF_DF + (cb + 3) * LD64 + h];
      t0 = t0 > 0.f ? t0 : 0.01f * t0;
      t1 = t1 > 0.f ? t1 : 0.01f * t1;
      t2 = t2 > 0.f ? t2 : 0.01f * t2;
      t3 = t3 > 0.f ? t3 : 0.01f * t3;
      a0 += t0 * wf; a1 += t1 * wf; a2 += t2 * wf; a3 += t3 * wf;
    }
    const size_t base = (size_t)b * 1024 + (size_t)a * 32 + cb;
    out_value[base + 0] = a0;
    out_value[base + 1] = a1;
    out_value[base + 2] = a2;
    out_value[base + 3] = a3;
  }
}

}  // namespace

extern "C" void kernel_launch(void* const* d_in, const int* in_sizes, int n_in,
                              void* d_out, int out_size, void* d_ws, size_t ws_size,
                              hipStream_t stream) {
  (void)in_sizes; (void)n_in; (void)d_ws; (void)ws_size; (void)out_size;
  const float* states   = (const float*)d_in[0];
  const float* policies = (const float*)d_in[1];
  const float* actions  = (const float*)d_in[2];
  const float* W_obs    = (const float*)d_in[3];
  const float* b_obs    = (const float*)d_in[4];
  const float* W_oa     = (const float*)d_in[5];
  const float* b_oa     = (const float*)d_in[6];
  const float* W_k      = (const float*)d_in[7];
  const float* b_k      = (const float*)d_in[8];
  const float* W_q      = (const float*)d_in[9];
  const float* b_q      = (const float*)d_in[10];
  const float* W_av     = (const float*)d_in[11];
  const float* W_f1     = (const float*)d_in[12];
  const float* W_f2     = (const float*)d_in[13];

  float* out_value  = (float*)d_out;
  float* out_weight = out_value + 256 * 32 * 32;  // value flat, then weight flat

  critic_fused<<<dim3(256), dim3(256), 0, stream>>>(
      states, policies, actions, W_obs, b_obs, W_oa, b_oa, W_k, b_k, W_q, b_q,
      W_av, W_f1, W_f2, out_value, out_weight);
}